// TrafficAugmentation_7567732375872
// MI455X (gfx1250) — compile-verified
//
#include <hip/hip_runtime.h>
#include <stdint.h>
#include <stddef.h>

// Problem constants (match reference).
#define NBATCH  4096
#define SEQ     512
#define OUTL    100
#define MSSV    1448
#define ROWS_WG 16
#define NWG     (NBATCH / ROWS_WG)   // 256 workgroups, 1 wave32 each
#define CSTRIDE 516                  // LDS row stride (16B aligned, bank-spread)

typedef float v2f __attribute__((ext_vector_type(2)));
typedef float v8f __attribute__((ext_vector_type(8)));
typedef int   v4i __attribute__((vector_size(16)));

#if defined(__gfx1250__) && __has_builtin(__builtin_amdgcn_wmma_f32_16x16x4_f32)
#define USE_WMMA 1
#else
#define USE_WMMA 0
#endif

#if defined(__gfx1250__) && __has_builtin(__builtin_amdgcn_global_load_async_to_lds_b128)
#define USE_ASYNC 1
typedef __attribute__((address_space(1))) v4i* glob_v4i_p;
typedef __attribute__((address_space(3))) v4i* lds_v4i_p;
#else
#define USE_ASYNC 0
#endif

struct SimWS {
  unsigned cnt;        // grid barrier arrival counter
  unsigned gen;        // grid barrier generation
  int      width[2];   // per-round batch-max packet count, double buffered by parity
  unsigned anyv[2];    // per-round "any valid remains", double buffered by parity
};

__global__ void ws_init_kernel(SimWS* ws) {
  if (threadIdx.x == 0) {
    ws->cnt = 0u; ws->gen = 0u;
    ws->width[0] = 0; ws->width[1] = 0;
    ws->anyv[0] = 0u; ws->anyv[1] = 0u;
  }
}

// Generation-counting device-wide barrier (all WGs co-resident: 256 WGs x 32 thr).
__device__ __forceinline__ void grid_barrier(SimWS* ws) {
  __threadfence();
  __syncthreads();
  if (threadIdx.x == 0) {
    unsigned g = __hip_atomic_load(&ws->gen, __ATOMIC_ACQUIRE, __HIP_MEMORY_SCOPE_AGENT);
    unsigned arrived =
        __hip_atomic_fetch_add(&ws->cnt, 1u, __ATOMIC_ACQ_REL, __HIP_MEMORY_SCOPE_AGENT) + 1u;
    if (arrived == (unsigned)NWG) {
      __hip_atomic_store(&ws->cnt, 0u, __ATOMIC_RELAXED, __HIP_MEMORY_SCOPE_AGENT);
      __hip_atomic_fetch_add(&ws->gen, 1u, __ATOMIC_RELEASE, __HIP_MEMORY_SCOPE_AGENT);
    } else {
      while (__hip_atomic_load(&ws->gen, __ATOMIC_ACQUIRE, __HIP_MEMORY_SCOPE_AGENT) == g) {
        __builtin_amdgcn_s_sleep(2);
      }
    }
  }
  __syncthreads();
  __threadfence();
}

// One wave32 per WG, 16 rows per WG.
// Lane mapping matches the V_WMMA_F32_16X16X4_F32 A-matrix layout:
//   lane l: row m = l&15; K-chunks c = {2*(l>>4), 2*(l>>4)+1}, each chunk = 128 seq elems.
__global__ __launch_bounds__(32)
void traffic_sim_kernel(const float* __restrict__ x,
                        const float* __restrict__ delays,
                        const float* __restrict__ rtts,
                        float* __restrict__ out,
                        SimWS* __restrict__ ws) {
  __shared__ float cumS[ROWS_WG][CSTRIDE];   // mutable per-element state, LDS-resident
  __shared__ float rttS[ROWS_WG];
  __shared__ float bufsS[ROWS_WG];
  __shared__ float partS[ROWS_WG][2];        // fallback reduction only
  __shared__ int   nS[ROWS_WG];
  __shared__ int   fullS[ROWS_WG];
  __shared__ int   remS[ROWS_WG];

  const int lane = threadIdx.x;
  const int m    = lane & 15;
  const int half = lane >> 4;
  const int g0   = blockIdx.x * ROWS_WG;
  const size_t rowbase = (size_t)(g0 + m) * SEQ;

  // ---- zero this WG's 16 output rows (exclusively owned) ----
  {
    const size_t ob = (size_t)g0 * OUTL;
    for (int t = lane; t < ROWS_WG * OUTL; t += 32) out[ob + t] = 0.0f;
  }

  // ---- per-lane exclusive valid bitmask: 8 x 32 bits = this lane's 256 elements ----
  unsigned valid[8];
  for (int w8 = 0; w8 < 8; ++w8) {
    const int sb = (half * 8 + w8) * 32;
    unsigned bits = 0u;
#pragma unroll
    for (int q = 0; q < 8; ++q) {
      const float4 xv = *reinterpret_cast<const float4*>(&x[rowbase + sb + q * 4]);
      const unsigned b = (unsigned)(q * 4);
      bits |= (xv.x > 0.0f ? 1u : 0u) << (b + 0);
      bits |= (xv.y > 0.0f ? 1u : 0u) << (b + 1);
      bits |= (xv.z > 0.0f ? 1u : 0u) << (b + 2);
      bits |= (xv.w > 0.0f ? 1u : 0u) << (b + 3);
    }
    valid[w8] = bits;
  }

  // ---- build cum0 (exclusive prefix sum of delays) in LDS ----
#if USE_ASYNC
  {
    // Async-stage delays rows into cumS (ASYNCcnt path), then in-place exclusive scan.
    const float* gsrc = delays + rowbase + half * 256;
    for (int it = 0; it < 64; ++it) {
      __builtin_amdgcn_global_load_async_to_lds_b128(
          (glob_v4i_p)(gsrc + it * 4),
          (lds_v4i_p)(&cumS[m][half * 256 + it * 4]),
          0, 0);
    }
#if __has_builtin(__builtin_amdgcn_s_wait_asynccnt)
    __builtin_amdgcn_s_wait_asynccnt(0);
#else
    asm volatile("s_wait_asynccnt 0" ::: "memory");
#endif
  }
  __syncthreads();
  if (half == 0) {
    float c = 0.0f;
    for (int s = 0; s < SEQ; ++s) {
      float d = cumS[m][s];
      cumS[m][s] = c;
      c += d;
    }
  }
#else
  if (half == 0) {
    float c = 0.0f;
    for (int s = 0; s < SEQ; ++s) {
      cumS[m][s] = c;
      c += delays[rowbase + s];
    }
  }
#endif
  __syncthreads();

  int offset = 0;  // replicated global scalar (identical arithmetic in every WG)

  for (int round = 0; round < SEQ; ++round) {
    const int p = round & 1;
    if (half == 0) rttS[m] = rtts[rowbase + round];
    __syncthreads();
    const float rtt_m = rttS[m];

    // ---- phase A: branchless window bits + masked row partial sums ----
    // float4 (b128) streams through LDS cum and L2-resident x; no per-element
    // dependent waits, compare+select only. Word-level skip drops drained words.
    unsigned wbits[8];
    float part0 = 0.0f, part1 = 0.0f;
#pragma unroll
    for (int j = 0; j < 2; ++j) {
      const int c = j + 2 * half;
      float sum = 0.0f;
#pragma unroll
      for (int w = 0; w < 4; ++w) {
        const unsigned vb = valid[j * 4 + w];
        unsigned wb = 0u;
        if (vb) {
          const int sb = c * 128 + w * 32;
#pragma unroll
          for (int q = 0; q < 8; ++q) {
            const int s = sb + q * 4;
            const float4 cv = *reinterpret_cast<const float4*>(&cumS[m][s]);
            const float4 xv = *reinterpret_cast<const float4*>(&x[rowbase + s]);
            const unsigned b = (unsigned)(q * 4);
            const unsigned b0 = ((vb >> (b + 0)) & 1u) & (cv.x < rtt_m ? 1u : 0u);
            const unsigned b1 = ((vb >> (b + 1)) & 1u) & (cv.y < rtt_m ? 1u : 0u);
            const unsigned b2 = ((vb >> (b + 2)) & 1u) & (cv.z < rtt_m ? 1u : 0u);
            const unsigned b3 = ((vb >> (b + 3)) & 1u) & (cv.w < rtt_m ? 1u : 0u);
            wb |= (b0 << (b + 0)) | (b1 << (b + 1)) | (b2 << (b + 2)) | (b3 << (b + 3));
            sum += b0 ? xv.x : 0.0f;     // exact: totals < 2^24 in f32
            sum += b1 ? xv.y : 0.0f;
            sum += b2 ? xv.z : 0.0f;
            sum += b3 ? xv.w : 0.0f;
          }
        }
        wbits[j * 4 + w] = wb;
      }
      if (j == 0) part0 = sum; else part1 = sum;
    }

    // ---- cross-lane reduction of 16 row sums via WMMA f32 16x16x4 (B = ones) ----
#if USE_WMMA
    {
      v2f a; a[0] = part0; a[1] = part1;               // A[m][k] per ISA 16x4 layout
      v2f bo; bo[0] = 1.0f; bo[1] = 1.0f;              // B = all ones -> D[m][*] = rowsum(m)
      v8f cz = {0.f, 0.f, 0.f, 0.f, 0.f, 0.f, 0.f, 0.f};
      v8f d = __builtin_amdgcn_wmma_f32_16x16x4_f32(
          false, a, false, bo, (short)0, cz, false, false);
      if (m == 0) {                                    // lanes 0 (rows 0-7) and 16 (rows 8-15)
#pragma unroll
        for (int v = 0; v < 8; ++v) bufsS[half * 8 + v] = d[v];
      }
    }
    __syncthreads();
#else
    partS[m][half] = part0 + part1;
    __syncthreads();
    if (half == 0) bufsS[m] = partS[m][0] + partS[m][1];
    __syncthreads();
#endif

    if (half == 0) {
      const int bi = (int)bufsS[m];
      const int fu = bi / MSSV;
      const int re = bi - fu * MSSV;
      fullS[m] = fu;
      remS[m]  = re;
      nS[m]    = fu + (re > 0 ? 1 : 0);
    }
    __syncthreads();

    if (lane == 0) {
      int nm = 0;
      for (int r = 0; r < ROWS_WG; ++r) nm = (nm > nS[r]) ? nm : nS[r];
      if (nm > 0) atomicMax(&ws->width[p], nm);
    }

    grid_barrier(ws);
    const int W = __hip_atomic_load(&ws->width[p], __ATOMIC_ACQUIRE, __HIP_MEMORY_SCOPE_AGENT);
    if (W == 0) break;   // uniform across grid

    if (lane == 0) {     // reset next round's (opposite parity) accumulators
      __hip_atomic_store(&ws->width[p ^ 1], 0, __ATOMIC_RELAXED, __HIP_MEMORY_SCOPE_AGENT);
      __hip_atomic_store(&ws->anyv[p ^ 1], 0u, __ATOMIC_RELAXED, __HIP_MEMORY_SCOPE_AGENT);
    }

    // ---- packet writes at the old (pre-advance) offset; disjoint columns per round ----
    {
      int kcap = OUTL - offset;
      if (kcap > W) kcap = W;
      if (kcap > 0) {
        for (int t = lane; t < ROWS_WG * kcap; t += 32) {
          const int r = t / kcap;
          const int k = t - r * kcap;
          if (k < nS[r]) {
            const float val = (k < fullS[r]) ? (float)(MSSV + 40) : (float)(remS[r] + 40);
            out[(size_t)(g0 + r) * OUTL + (offset + k)] = val;
          }
        }
      }
    }

    // ---- state update: valid &= ~window; cum += delays where still valid & !window ----
    // Branchless float4 read-modify-write; each lane exclusively owns its words, so
    // writing back unchanged values for masked-off elements is safe.
    unsigned remain = 0u;
#pragma unroll
    for (int j = 0; j < 2; ++j) {
      const int c = j + 2 * half;
#pragma unroll
      for (int w = 0; w < 4; ++w) {
        const unsigned nv = valid[j * 4 + w] & ~wbits[j * 4 + w];
        valid[j * 4 + w] = nv;
        remain |= nv;
        if (nv) {
          const int sb = c * 128 + w * 32;
#pragma unroll
          for (int q = 0; q < 8; ++q) {
            const int s = sb + q * 4;
            const float4 dv = *reinterpret_cast<const float4*>(&delays[rowbase + s]);
            float4 cv = *reinterpret_cast<float4*>(&cumS[m][s]);
            const unsigned b = (unsigned)(q * 4);
            cv.x += ((nv >> (b + 0)) & 1u) ? dv.x : 0.0f;
            cv.y += ((nv >> (b + 1)) & 1u) ? dv.y : 0.0f;
            cv.z += ((nv >> (b + 2)) & 1u) ? dv.z : 0.0f;
            cv.w += ((nv >> (b + 3)) & 1u) ? dv.w : 0.0f;
            *reinterpret_cast<float4*>(&cumS[m][s]) = cv;
          }
        }
      }
    }
    if (remain) atomicOr(&ws->anyv[p], 1u);

    offset += W;

    grid_barrier(ws);
    const unsigned ga =
        __hip_atomic_load(&ws->anyv[p], __ATOMIC_ACQUIRE, __HIP_MEMORY_SCOPE_AGENT);
    if (ga == 0u) break;       // reference: done when no valid packets remain
    if (offset >= OUTL) break; // output frozen past column 100 -> identical result
  }
}

extern "C" void kernel_launch(void* const* d_in, const int* in_sizes, int n_in,
                              void* d_out, int out_size, void* d_ws, size_t ws_size,
                              hipStream_t stream) {
  (void)in_sizes; (void)n_in; (void)out_size; (void)ws_size;
  const float* x      = (const float*)d_in[0];
  const float* delays = (const float*)d_in[1];
  const float* rtts   = (const float*)d_in[2];
  float* out = (float*)d_out;
  SimWS* ws  = (SimWS*)d_ws;

  hipLaunchKernelGGL(ws_init_kernel, dim3(1), dim3(32), 0, stream, ws);
  hipLaunchKernelGGL(traffic_sim_kernel, dim3(NWG), dim3(32), 0, stream,
                     x, delays, rtts, out, ws);
}